// SparseTopKAttention_56538949484885
// MI455X (gfx1250) — compile-verified
//
#include <hip/hip_runtime.h>
#include <hip/hip_bf16.h>
#include <math.h>

// Problem constants (match reference)
#define B_   4
#define NQ_  4096
#define NK_  4096
#define D_   256
#define H_   8
#define DH_  32
#define K_   32
#define SCALE_ 0.17677669529663687f   // 1/sqrt(32)

typedef float v2f __attribute__((ext_vector_type(2)));
typedef float v8f __attribute__((ext_vector_type(8)));

__device__ __forceinline__ void wait_asynccnt0() {
#if __has_builtin(__builtin_amdgcn_s_wait_asynccnt)
    __builtin_amdgcn_s_wait_asynccnt(0);
#else
    asm volatile("s_wait_asynccnt 0x0" ::: "memory");
#endif
}

// ---------------------------------------------------------------------------
// GEMM: C[M x 256] = A[M x 256] @ W[256 x 256] + bias, f32 WMMA 16x16x4.
// Block = 8 waves, all sharing one 16-column W tile (tn), distinct 16-row
// A tiles (tm). The 16KB W tile is staged into LDS with the CDNA5 async
// global->LDS copy path (ASYNCcnt), then B-fragments come from conflict-free
// ds_load broadcasts while A-fragments stream from global.
//   A-frag  (16x4): lane l -> row l&15, holds K = 2*(l>>4), 2*(l>>4)+1
//   B-frag  (4x16): lane l -> col l&15, holds K = 2*(l>>4), 2*(l>>4)+1
//   C-frag (16x16): VGPR i, lane l -> row i + 8*(l>>4), col l&15
// ---------------------------------------------------------------------------
__global__ __launch_bounds__(256) void gemm256_wmma_f32(
    const float* __restrict__ A, const float* __restrict__ W,
    const float* __restrict__ bias, float* __restrict__ C, int M)
{
    __shared__ float s_w[D_][16];               // 16 KB W tile: s_w[k][c]

    const int lane = threadIdx.x & 31;
    const int wave = threadIdx.x >> 5;
    const int tn   = blockIdx.x & 15;           // waves share W columns
    const int tm   = (blockIdx.x >> 4) * 8 + wave;

    // ---- stage W[:, tn*16 .. tn*16+15] into LDS via async copy (16B/lane) --
    {
        const float* __restrict__ wsrc = W + tn * 16;
        #pragma unroll
        for (int it = 0; it < 4; ++it) {
            const int i   = threadIdx.x + it * 256;   // 0..1023 float4 chunks
            const int row = i >> 2;                   // 0..255 (k)
            const int c4  = (i & 3) << 2;             // 0,4,8,12
            const unsigned lds_off = (unsigned)(size_t)(&s_w[row][c4]);
            const float*   g       = wsrc + (size_t)row * D_ + c4;
            asm volatile("global_load_async_to_lds_b128 %0, %1, off"
                         :: "v"(lds_off), "v"(g) : "memory");
        }
        wait_asynccnt0();
    }
    __syncthreads();

    const int c    = lane & 15;                 // column within tile
    const int ncol = tn * 16 + c;               // global C/B column
    const int kgrp = (lane >> 4) << 1;          // K-pair base for this lane

    v8f acc;
    const float bval = bias[ncol];
    #pragma unroll
    for (int i = 0; i < 8; ++i) acc[i] = bval;

    const float* __restrict__ arow = A + (size_t)(tm * 16 + c) * D_;

    #pragma unroll 4
    for (int k = 0; k < D_; k += 4) {
        v2f a, b;
        a[0] = arow[k + kgrp];
        a[1] = arow[k + kgrp + 1];
        b[0] = s_w[k + kgrp][c];
        b[1] = s_w[k + kgrp + 1][c];
        acc = __builtin_amdgcn_wmma_f32_16x16x4_f32(
            /*neg_a=*/false, a, /*neg_b=*/false, b,
            /*c_mod=*/(short)0, acc, /*reuse_a=*/false, /*reuse_b=*/false);
    }

    const int rbase = tm * 16 + ((lane >> 4) << 3);
    #pragma unroll
    for (int i = 0; i < 8; ++i)
        C[(size_t)(rbase + i) * D_ + ncol] = acc[i];
}

// ---------------------------------------------------------------------------
// Rotary encode, in place. One thread per (row, head, pair j<16).
// Even heads: 2D RoPE from pos; odd heads: scalar heading rotation.
// ---------------------------------------------------------------------------
__global__ void encode_rope(float* __restrict__ X, const float* __restrict__ pos,
                            const float* __restrict__ heading, int nrows)
{
    const int gid = blockIdx.x * blockDim.x + threadIdx.x;
    if (gid >= nrows * 128) return;
    const int row = gid >> 7;
    const int t   = gid & 127;
    const int h   = t >> 4;         // head 0..7
    const int j   = t & 15;         // pair index within half (0..15)

    float c, s;
    if ((h & 1) == 0) {
        const float px = pos[(size_t)row * 2 + 0];
        const float py = pos[(size_t)row * 2 + 1];
        const int   f  = j & 7;
        const float freq = powf(10000.0f, -(float)f * 0.125f);
        const float ang  = (j < 8 ? px : py) * freq;
        sincosf(ang, &s, &c);
    } else {
        sincosf(heading[row], &s, &c);
    }

    float* __restrict__ xr = X + (size_t)row * D_ + h * DH_;
    const float v1 = xr[j];
    const float v2 = xr[j + 16];
    xr[j]      = v1 * c - v2 * s;
    xr[j + 16] = v2 * c + v1 * s;
}

// ---------------------------------------------------------------------------
// Top-32 nearest keys per query by squared 2D distance.
// 128 threads/block, one query per thread. pos_k for the batch sits in LDS as
// interleaved float2 (32 KB of the WGP's 320 KB); every thread in the block
// reads the SAME key index each iteration -> single broadcast ds_load_b64.
// Top-32 is a register compare-insert chain, entered only on improvement.
// ---------------------------------------------------------------------------
__global__ __launch_bounds__(128) void topk_kernel(
    const float* __restrict__ pos_q, const float* __restrict__ pos_k,
    const unsigned char* __restrict__ mask_k, int* __restrict__ topk_idx)
{
    __shared__ float2 s_kp[NK_];                // 32 KB

    const int b      = blockIdx.x >> 5;         // 32 blocks per batch
    const int qlocal = ((blockIdx.x & 31) << 7) + threadIdx.x;
    const int q      = b * NQ_ + qlocal;

    const float*         pk = pos_k  + (size_t)b * NK_ * 2;
    const unsigned char* mk = mask_k + (size_t)b * NK_;
    const float2 far2 = make_float2(3.0e38f, 3.0e38f);
    for (int i = threadIdx.x; i < NK_; i += 128) {
        const float2 p = ((const float2*)pk)[i];
        s_kp[i] = (mk[i] != 0) ? p : far2;      // masked -> effectively +inf
    }
    __syncthreads();

    const float qx = pos_q[(size_t)q * 2 + 0];
    const float qy = pos_q[(size_t)q * 2 + 1];

    float bd[K_]; int bi[K_];
    #pragma unroll
    for (int i = 0; i < K_; ++i) { bd[i] = 3.0e38f; bi[i] = i; }

    for (int k = 0; k < NK_; k += 4) {
        #pragma unroll
        for (int u = 0; u < 4; ++u) {
            const float2 p  = s_kp[k + u];      // broadcast across the wave
            const float  dx = qx - p.x;
            const float  dy = qy - p.y;
            const float  d2 = dx * dx + dy * dy;
            if (d2 < bd[K_ - 1]) {
                float cd = d2; int ci = k + u;
                #pragma unroll
                for (int i = 0; i < K_; ++i) {
                    const bool  sw = cd < bd[i];
                    const float nd = sw ? cd : bd[i];
                    const int   ni = sw ? ci : bi[i];
                    cd = sw ? bd[i] : cd;
                    ci = sw ? bi[i] : ci;
                    bd[i] = nd; bi[i] = ni;
                }
            }
        }
    }

    int* __restrict__ out = topk_idx + (size_t)q * K_;
    #pragma unroll
    for (int i = 0; i < K_; ++i) out[i] = bi[i];
}

// ---------------------------------------------------------------------------
// Attention: one block per query, one wave per head (wave32).
// Phase 1: lane = key   -> dot(Q_h, K_h[key]) via float4 loads + q shuffles.
// Phase 2: softmax across the 32 lanes with __shfl_xor.
// Phase 3: lane = dim d -> sum_k attn[k] * V[key_k][d] (coalesced 128B reads).
// ---------------------------------------------------------------------------
__global__ __launch_bounds__(256) void attn_kernel(
    const float* __restrict__ Qenc, const float* __restrict__ Kenc,
    const float* __restrict__ Vbuf, const int* __restrict__ topk_idx,
    const unsigned char* __restrict__ mask_k, float* __restrict__ Out)
{
    const int q    = blockIdx.x;            // 0..B*NQ-1
    const int b    = q >> 12;               // / 4096
    const int h    = threadIdx.x >> 5;      // wave == head
    const int lane = threadIdx.x & 31;

    const int    key   = topk_idx[(size_t)q * K_ + lane];
    const size_t kbase = ((size_t)b * NK_ + key) * D_ + h * DH_;
    const size_t qbase = (size_t)q * D_ + h * DH_;

    const float qreg = Qenc[qbase + lane];  // lane holds one dim of Q_h

    float dot = 0.0f;
    #pragma unroll
    for (int j = 0; j < 8; ++j) {
        const float4 kv = *(const float4*)(Kenc + kbase + j * 4);
        dot += kv.x * __shfl(qreg, j * 4 + 0, 32);
        dot += kv.y * __shfl(qreg, j * 4 + 1, 32);
        dot += kv.z * __shfl(qreg, j * 4 + 2, 32);
        dot += kv.w * __shfl(qreg, j * 4 + 3, 32);
    }

    const bool valid = mask_k[(size_t)b * NK_ + key] != 0;
    float logit = valid ? dot * SCALE_ : -__builtin_inff();

    float m = logit;
    #pragma unroll
    for (int off = 16; off > 0; off >>= 1)
        m = fmaxf(m, __shfl_xor(m, off, 32));
    const float e = expf(logit - m);
    float ssum = e;
    #pragma unroll
    for (int off = 16; off > 0; off >>= 1)
        ssum += __shfl_xor(ssum, off, 32);
    const float attn = e / ssum;

    float acc = 0.0f;
    #pragma unroll 4
    for (int k = 0; k < K_; ++k) {
        const float a  = __shfl(attn, k, 32);
        const int   ki = __shfl(key,  k, 32);
        acc += a * Vbuf[((size_t)b * NK_ + ki) * D_ + h * DH_ + lane];
    }
    Out[qbase + lane] = acc;
}

// ---------------------------------------------------------------------------
// Launch: three projection GEMMs -> rotary encodes -> topk -> attention ->
// output GEMM into d_out. All on `stream`; workspace laid out below (~66 MB).
// ---------------------------------------------------------------------------
extern "C" void kernel_launch(void* const* d_in, const int* in_sizes, int n_in,
                              void* d_out, int out_size, void* d_ws, size_t ws_size,
                              hipStream_t stream)
{
    const float* q_feat    = (const float*)d_in[0];
    const float* kv_feat   = (const float*)d_in[1];
    const float* pos_q     = (const float*)d_in[2];
    const float* pos_k     = (const float*)d_in[3];
    const float* heading_q = (const float*)d_in[4];
    const float* heading_k = (const float*)d_in[5];
    const unsigned char* mask_k = (const unsigned char*)d_in[6];
    const float* Wq = (const float*)d_in[7];
    const float* bq = (const float*)d_in[8];
    const float* Wk = (const float*)d_in[9];
    const float* bk = (const float*)d_in[10];
    const float* Wv = (const float*)d_in[11];
    const float* bv = (const float*)d_in[12];
    const float* Wo = (const float*)d_in[13];
    const float* bo = (const float*)d_in[14];

    const int Mq = B_ * NQ_;    // 16384 query rows
    const int Mk = B_ * NK_;    // 16384 key rows
    const size_t rowElems = (size_t)Mq * D_;   // 4 Mi floats per buffer

    float* Qenc = (float*)d_ws;
    float* Kenc = Qenc + rowElems;
    float* Vbuf = Kenc + rowElems;
    float* Aout = Vbuf + rowElems;
    int*   idx  = (int*)(Aout + rowElems);     // B*NQ*32 ints

    const int gemmBlocks = (Mq / 128) * 16;    // 2048: (tm groups) x (tn)

    // 1) projections
    gemm256_wmma_f32<<<gemmBlocks, 256, 0, stream>>>(q_feat,  Wq, bq, Qenc, Mq);
    gemm256_wmma_f32<<<gemmBlocks, 256, 0, stream>>>(kv_feat, Wk, bk, Kenc, Mk);
    gemm256_wmma_f32<<<gemmBlocks, 256, 0, stream>>>(kv_feat, Wv, bv, Vbuf, Mk);

    // 2) rotary encodes (in place)
    const int encBlocksQ = (Mq * 128 + 255) / 256;
    encode_rope<<<encBlocksQ, 256, 0, stream>>>(Qenc, pos_q, heading_q, Mq);
    encode_rope<<<encBlocksQ, 256, 0, stream>>>(Kenc, pos_k, heading_k, Mk);

    // 3) top-32 neighbors per query
    topk_kernel<<<B_ * (NQ_ / 128), 128, 0, stream>>>(pos_q, pos_k, mask_k, idx);

    // 4) gathered attention
    attn_kernel<<<Mq, 256, 0, stream>>>(Qenc, Kenc, Vbuf, idx, mask_k, Aout);

    // 5) output projection into d_out
    gemm256_wmma_f32<<<gemmBlocks, 256, 0, stream>>>(Aout, Wo, bo, (float*)d_out, Mq);
}